// SIFTNet_91087666413577
// MI455X (gfx1250) — compile-verified
//
#include <hip/hip_runtime.h>
#include <math.h>

typedef float v2f __attribute__((ext_vector_type(2)));
typedef float v8f __attribute__((ext_vector_type(8)));

#define H    4096
#define WD   4096
#define OH   4097
#define OW   4097
#define NCH  8

#define BY   32          // output rows per block
#define BX   32          // output cols per block
#define NTHR 64          // 2 wave32s

#define INROWS 38        // input halo rows: y0-3 .. y0+34
#define INCOLS 40        // 38 used, padded
#define HRN    36        // hist halo rows: y0-2 .. y0+33
#define HCN    36        // hist halo cols: x0-2 .. x0+33
#define VCP    40        // V col pad (36 used)

// cos/sin of (2i+1)*pi/8
#define CA 0.92387953251128674f   // cos 22.5
#define CB 0.38268343236508978f   // cos 67.5

__global__ __launch_bounds__(NTHR)
void sift_fused_kernel(const float* __restrict__ xin, float* __restrict__ out)
{
    __shared__ __align__(16) float s_in[INROWS][INCOLS];
    __shared__ __align__(16) float s_V[NCH][BY][VCP];   // vertical 4-tap box sums, per channel

    const int tid = threadIdx.x;
    const int y0  = blockIdx.y * BY;
    const int x0  = blockIdx.x * BX;

    // block-uniform interior tests (SGPR-resolvable)
    const bool halo_interior  = (y0 >= 3) && (y0 + 34 < H) && (x0 >= 3) && (x0 + 34 < WD);
    const bool store_interior = (y0 + BY <= OH) && (x0 + BX <= OW);

    // ---- zero V (LDS is not zeroed by HW) ----
    {
        float4* vp = (float4*)&s_V[0][0][0];
        const float4 z = {0.f, 0.f, 0.f, 0.f};
        const int n4 = (NCH * BY * VCP) / 4;
        for (int i = tid; i < n4; i += NTHR) vp[i] = z;
    }

    // ---- stage input halo tile into LDS (zero padding == conv padding) ----
    if (halo_interior) {
        const float* base = xin + (y0 - 3) * WD + (x0 - 3);
        for (int i = tid; i < INROWS * INCOLS; i += NTHR) {
            int r = i / INCOLS, c = i - r * INCOLS;
            s_in[r][c] = base[r * WD + c];
        }
    } else {
        for (int i = tid; i < INROWS * INCOLS; i += NTHR) {
            int r = i / INCOLS, c = i - r * INCOLS;
            int gy = y0 - 3 + r, gx = x0 - 3 + c;
            float v = 0.f;
            if ((unsigned)gy < (unsigned)H && (unsigned)gx < (unsigned)WD)
                v = xin[gy * WD + gx];
            s_in[r][c] = v;
        }
    }
    __syncthreads();

    // ---- Sobel + orientation argmax + sparse scatter into vertical sums ----
    for (int i = tid; i < HRN * HCN; i += NTHR) {
        int hr = i / HCN, hc = i - hr * HCN;        // hist pixel (y0-2+hr, x0-2+hc)
        int gy = y0 - 2 + hr, gx = x0 - 2 + hc;
        if (halo_interior ||
            ((unsigned)gy < (unsigned)H && (unsigned)gx < (unsigned)WD)) {
            float i00 = s_in[hr    ][hc], i01 = s_in[hr    ][hc + 1], i02 = s_in[hr    ][hc + 2];
            float i10 = s_in[hr + 1][hc],                             i12 = s_in[hr + 1][hc + 2];
            float i20 = s_in[hr + 2][hc], i21 = s_in[hr + 2][hc + 1], i22 = s_in[hr + 2][hc + 2];
            float dx = (i02 - i00) + 2.f * (i12 - i10) + (i22 - i20);
            float dy = (i20 - i00) + 2.f * (i21 - i01) + (i22 - i02);
            float mag = sqrtf(dx * dx + dy * dy);

            const float CS[8] = { CA,  CB, -CB, -CA, -CA, -CB,  CB,  CA };
            const float SN[8] = { CB,  CA,  CA,  CB, -CB, -CA, -CA, -CB };
            int best = 0;
            float bv = CS[0] * dx + SN[0] * dy;
            #pragma unroll
            for (int k = 1; k < 8; ++k) {
                float v = CS[k] * dx + SN[k] * dy;
                if (v > bv) { bv = v; best = k; }   // first-max semantics like jnp.argmax
            }
            // hist row hr contributes to output rows (block-local) hr-3 .. hr
            #pragma unroll
            for (int k = 0; k < 4; ++k) {
                int y = hr - 3 + k;
                if ((unsigned)y < (unsigned)BY)
                    atomicAdd(&s_V[best][y][hc], mag);   // ds_add_f32, one-hot channel
            }
        }
    }
    __syncthreads();

    // ---- horizontal 4-tap box sum via V_WMMA_F32_16X16X4_F32 ----
    const int lane = tid & 31;
    const int wv   = tid >> 5;                 // wave id: rows 16*wv .. 16*wv+15
    const int m    = lane & 15;                // M (A) == N (B/C) index
    const int k0   = (lane < 16) ? 0 : 2;      // K pair held by this lane half

    // constant banded selection matrices: S_j[k][n] = 1 iff n in [4j+k-3, 4j+k]
    v2f Bsel[5];
    #pragma unroll
    for (int jj = 0; jj < 5; ++jj) {
        int t0 = 4 * jj + k0;
        Bsel[jj].x = (m <= t0     && m >= t0 - 3) ? 1.f : 0.f;
        Bsel[jj].y = (m <= t0 + 1 && m >= t0 - 2) ? 1.f : 0.f;
    }

    const int vrow   = m + 16 * wv;
    const int rowadd = (lane < 16) ? 0 : 8;

    for (int c = 0; c < NCH; ++c) {
        #pragma unroll
        for (int t = 0; t < 2; ++t) {          // two 16-wide output tiles
            v8f acc = {0.f, 0.f, 0.f, 0.f, 0.f, 0.f, 0.f, 0.f};
            #pragma unroll
            for (int jj = 0; jj < 5; ++jj) {
                int vcb = 16 * t + 4 * jj + k0;                  // even -> 8B aligned
                v2f a = *(const v2f*)&s_V[c][vrow][vcb];         // ds_load_b64
                acc = __builtin_amdgcn_wmma_f32_16x16x4_f32(
                        false, a, false, Bsel[jj], (short)0, acc, false, false);
            }
            // C layout: VGPR r, lanes 0-15 -> row r; lanes 16-31 -> row r+8; col = lane%16
            const int  ocol  = x0 + 16 * t + m;
            const int  orow  = y0 + 16 * wv + rowadd;
            const long cbase = (long)c * ((long)OH * OW);
            if (store_interior) {
                // fast path: whole tile in-bounds, straight-line coalesced stores
                float* p = out + cbase + (long)orow * OW + ocol;
                #pragma unroll
                for (int r = 0; r < 8; ++r)
                    p[(long)r * OW] = acc[r];
            } else if (ocol < OW) {
                #pragma unroll
                for (int r = 0; r < 8; ++r) {
                    int gy = orow + r;
                    if (gy < OH)
                        out[cbase + (long)gy * OW + ocol] = acc[r];
                }
            }
        }
    }
}

extern "C" void kernel_launch(void* const* d_in, const int* in_sizes, int n_in,
                              void* d_out, int out_size, void* d_ws, size_t ws_size,
                              hipStream_t stream)
{
    const float* x = (const float*)d_in[0];
    float* out = (float*)d_out;
    dim3 grid((OW + BX - 1) / BX, (OH + BY - 1) / BY);   // 129 x 129
    sift_fused_kernel<<<grid, dim3(NTHR), 0, stream>>>(x, out);
}